// ClsCrossAttention_5935644803511
// MI455X (gfx1250) — compile-verified
//
#include <hip/hip_runtime.h>

typedef float v2f __attribute__((ext_vector_type(2)));
typedef float v8f __attribute__((ext_vector_type(8)));

#define B_  64
#define C_  768
#define N_  1024
#define H_  12
#define HP  16      // heads padded to 16 for WMMA
#define DK  64
#define LDW 68      // padded LDS row (floats): bank-conflict free for row-indexed lanes

// ---------------------------------------------------------------------------
// Kernel 1: fold cls into W_k and pos_embed (tiny).
//   wT[c][h] = (1/8) * sum_d W_k[h,c,d]*cls[h,d]   (h>=12 -> 0)
//   pT[n][h] = (1/8) * sum_d pos[n,d] *cls[h,d]    (h>=12 -> 0)
// ---------------------------------------------------------------------------
__global__ __launch_bounds__(256) void cls_fold_kernel(
    const float* __restrict__ cls, const float* __restrict__ W_k,
    const float* __restrict__ pos, float* __restrict__ wT, float* __restrict__ pT) {
  int idx = blockIdx.x * blockDim.x + threadIdx.x;
  const float scale = 0.125f;   // 1/sqrt(64)
  if (idx < C_ * HP) {
    int c = idx >> 4, h = idx & 15;
    float s = 0.f;
    if (h < H_) {
      const float* wk = W_k + ((size_t)h * C_ + c) * DK;
      const float* cl = cls + h * DK;
      #pragma unroll 8
      for (int d = 0; d < DK; ++d) s += wk[d] * cl[d];
    }
    wT[c * HP + h] = s * scale;
  } else if (idx < C_ * HP + N_ * HP) {
    int j = idx - C_ * HP;
    int n = j >> 4, h = j & 15;
    float s = 0.f;
    if (h < H_) {
      const float* pe = pos + (size_t)n * DK;
      const float* cl = cls + h * DK;
      #pragma unroll 8
      for (int d = 0; d < DK; ++d) s += pe[d] * cl[d];
    }
    pT[n * HP + h] = s * scale;
  }
}

// ---------------------------------------------------------------------------
// Kernel 2: logits[b][n][h16] = (tokens @ wT) + pT via V_WMMA_F32_16X16X4_F32.
// One wave per TWO adjacent 16-n tiles (32-n span): the B fragment (wT) is
// loaded once and feeds two WMMAs, and the paired A loads cover full 128B
// cachelines of features[b][c][n].
// ---------------------------------------------------------------------------
__global__ __launch_bounds__(128) void logits_kernel(
    const float* __restrict__ feat, const float* __restrict__ wT,
    const float* __restrict__ pT, float* __restrict__ logitsT) {
  const int lane = threadIdx.x & 31;
  const int wave = threadIdx.x >> 5;
  const int half = lane >> 4;     // 0: K pair {0,1}, 1: K pair {2,3}
  const int l16  = lane & 15;
  const int b  = blockIdx.y;
  const int n0 = (blockIdx.x * 4 + wave) * 32;
  const float* F = feat + (size_t)b * C_ * N_;

  v8f acc0 = {}, acc1 = {};
  for (int c0 = 0; c0 < C_; c0 += 4) {
    const int ca = c0 + 2 * half;
    const float* r0 = F + (size_t)(ca + 0) * N_ + l16;
    const float* r1 = F + (size_t)(ca + 1) * N_ + l16;
    v2f Bm = { wT[(ca + 0) * HP + l16], wT[(ca + 1) * HP + l16] };
    v2f A0 = { r0[n0],      r1[n0] };
    v2f A1 = { r0[n0 + 16], r1[n0 + 16] };
    acc0 = __builtin_amdgcn_wmma_f32_16x16x4_f32(
        false, A0, false, Bm, (short)0, acc0, false, false);
    acc1 = __builtin_amdgcn_wmma_f32_16x16x4_f32(
        false, A1, false, Bm, (short)0, acc1, false, false);
  }
  float* out = logitsT + (size_t)b * N_ * HP;
  #pragma unroll
  for (int r = 0; r < 8; ++r) {
    const int n = n0 + r + 8 * half;   // D layout: VGPR r = row r (+8 for hi lanes)
    out[n * HP + l16]        = acc0[r] + pT[n * HP + l16];
    out[(n + 16) * HP + l16] = acc1[r] + pT[(n + 16) * HP + l16];
  }
}

// ---------------------------------------------------------------------------
// Kernel 3: softmax over n per (b,h), in place on [b][n][16h] layout.
// Block per b; thread (s,h) owns a 64-n stripe of head h, LDS tree-reduce.
// ---------------------------------------------------------------------------
__global__ __launch_bounds__(256) void softmax_kernel(float* __restrict__ att) {
  const int b = blockIdx.x;
  const int tid = threadIdx.x;
  const int h = tid & 15;
  const int s = tid >> 4;          // 16 stripes of 64 tokens
  float* base = att + (size_t)b * N_ * HP;

  float v[64];
  float mx = -3.4e38f;
  #pragma unroll
  for (int i = 0; i < 64; ++i) {
    v[i] = base[(s * 64 + i) * HP + h];
    mx = fmaxf(mx, v[i]);
  }
  __shared__ float red[256];
  red[tid] = mx;
  __syncthreads();
  #pragma unroll
  for (int off = 8; off >= 1; off >>= 1) {
    if (s < off) red[tid] = fmaxf(red[tid], red[tid + off * 16]);
    __syncthreads();
  }
  mx = red[h];
  __syncthreads();

  float sum = 0.f;
  #pragma unroll
  for (int i = 0; i < 64; ++i) {
    v[i] = __expf(v[i] - mx);
    sum += v[i];
  }
  red[tid] = sum;
  __syncthreads();
  #pragma unroll
  for (int off = 8; off >= 1; off >>= 1) {
    if (s < off) red[tid] += red[tid + off * 16];
    __syncthreads();
  }
  const float inv = 1.0f / red[h];
  #pragma unroll
  for (int i = 0; i < 64; ++i)
    base[(s * 64 + i) * HP + h] = v[i] * inv;
}

// ---------------------------------------------------------------------------
// Kernel 4: out[b][h][c] = sum_n attn[b][n][h] * features[b][c][n] via WMMA.
// Wave computes TWO D[16c x 16h] tiles (32-c span): features tile 32c x 64n
// staged through padded LDS (coalesced float4 global loads, conflict-free
// ds reads), attn B fragment loaded once per K-step feeding both WMMAs.
// ---------------------------------------------------------------------------
__global__ __launch_bounds__(128) void pool_kernel(
    const float* __restrict__ feat, const float* __restrict__ att,
    float* __restrict__ out) {
  __shared__ float lds[4][32 * LDW];
  const int lane = threadIdx.x & 31;
  const int wave = threadIdx.x >> 5;
  const int half = lane >> 4;
  const int l16  = lane & 15;
  const int b  = blockIdx.y;
  const int c0 = (blockIdx.x * 4 + wave) * 32;
  const float* F = feat + (size_t)b * C_ * N_;
  const float* A = att + (size_t)b * N_ * HP;
  float* ldsw = lds[wave];

  v8f acc0 = {}, acc1 = {};
  for (int nc = 0; nc < N_; nc += 64) {
    // Stage 32c x 64n tile: 16 float4 per lane, coalesced 256B per row.
    #pragma unroll
    for (int i = 0; i < 16; ++i) {
      const int idx = i * 32 + lane;
      const int row = idx >> 4;          // 16 float4 per row
      const int col = (idx & 15) * 4;
      const float4 t = *(const float4*)&F[(size_t)(c0 + row) * N_ + nc + col];
      float* dst = &ldsw[row * LDW + col];
      dst[0] = t.x; dst[1] = t.y; dst[2] = t.z; dst[3] = t.w;
    }
    __syncthreads();
    #pragma unroll
    for (int k0 = 0; k0 < 64; k0 += 4) {
      const int ka = k0 + 2 * half;
      v2f Bf = { A[(nc + ka + 0) * HP + l16],
                 A[(nc + ka + 1) * HP + l16] };
      v2f A0 = { ldsw[l16 * LDW + ka + 0],
                 ldsw[l16 * LDW + ka + 1] };
      v2f A1 = { ldsw[(l16 + 16) * LDW + ka + 0],
                 ldsw[(l16 + 16) * LDW + ka + 1] };
      acc0 = __builtin_amdgcn_wmma_f32_16x16x4_f32(
          false, A0, false, Bf, (short)0, acc0, false, false);
      acc1 = __builtin_amdgcn_wmma_f32_16x16x4_f32(
          false, A1, false, Bf, (short)0, acc1, false, false);
    }
    __syncthreads();
  }
  #pragma unroll
  for (int r = 0; r < 8; ++r) {
    const int c = c0 + r + 8 * half;
    if (l16 < H_) {
      out[(size_t)b * (H_ * C_) + l16 * C_ + c]      = acc0[r];
      out[(size_t)b * (H_ * C_) + l16 * C_ + c + 16] = acc1[r];
    }
  }
}

// ---------------------------------------------------------------------------
extern "C" void kernel_launch(void* const* d_in, const int* in_sizes, int n_in,
                              void* d_out, int out_size, void* d_ws, size_t ws_size,
                              hipStream_t stream) {
  const float* features = (const float*)d_in[0];   // [64,768,32,32]
  const float* cls      = (const float*)d_in[1];   // [12,64]
  const float* W_k      = (const float*)d_in[2];   // [12,768,64]
  const float* pos      = (const float*)d_in[3];   // [1,1,1024,64]
  float* out = (float*)d_out;                       // [64, 9216]

  char* ws = (char*)d_ws;
  float* wT      = (float*)ws;                               // 768*16*4  = 48 KB
  float* pT      = (float*)(ws + 49152);                     // 1024*16*4 = 64 KB
  float* logitsT = (float*)(ws + 49152 + 65536);             // 64*1024*16*4 = 4 MB

  cls_fold_kernel<<<dim3((C_ * HP + N_ * HP + 255) / 256), 256, 0, stream>>>(
      cls, W_k, pos, wT, pT);
  logits_kernel<<<dim3(N_ / 32 / 4, B_), 128, 0, stream>>>(
      features, wT, pT, logitsT);
  softmax_kernel<<<dim3(B_), 256, 0, stream>>>(logitsT);
  pool_kernel<<<dim3(C_ / 32 / 4, B_), 128, 0, stream>>>(
      features, logitsT, out);
}